// StructureBiasSineNonStationaryRPE_85693187490166
// MI455X (gfx1250) — compile-verified
//
#include <hip/hip_runtime.h>
#include <hip/hip_bf16.h>

#define B_ 2
#define S_ 384
#define D_ 128

typedef __attribute__((ext_vector_type(2))) float v2f;
typedef __attribute__((ext_vector_type(4))) float v4f;
typedef __attribute__((ext_vector_type(8))) float v8f;

// ---- table row stacking (all multiples of 16 rows) ----
// rows [0,48)    tabA: chord RPE,      v = row - 23        (valid v in [-23,23])
// rows [48,80)   tabB: ann RPE,        v = row - 48 - 15   (valid v in [-15,15])
// rows [80,848)  tabC: chord nonstat,  v = row - 80 - 383
// rows [848,1616)tabD: ann nonstat,    v = row - 848 - 383
#define ROWS_A 48
#define ROWS_B 32
#define ROWS_C 768
#define ROWS_D 768
#define ROWS_TOTAL (ROWS_A + ROWS_B + ROWS_C + ROWS_D)   // 1616

// ---- workspace layout (floats) ----
#define TABA_OFF 0
#define TABB_OFF (ROWS_A * 128)
#define TABC_OFF (TABB_OFF + ROWS_B * 128)
#define TABD_OFF (TABC_OFF + ROWS_C * 128)
#define TABLES_FLOATS (ROWS_TOTAL * 128)
#define ENC_OFF TABLES_FLOATS                    // enc: ROWS_TOTAL x 128
#define INTS_OFF (TABLES_FLOATS + ROWS_TOTAL * 128)

// ---------------------------------------------------------------------------
// Phase 0: per-batch segment-relative time for chord & annotation sequences.
// t[s] = s - start_of_current_segment. Serial scan, 4 tiny tasks.
// ---------------------------------------------------------------------------
__global__ void phase0_times(const int* __restrict__ chord,
                             const int* __restrict__ ann,
                             int* __restrict__ tC, int* __restrict__ tA) {
  int task = threadIdx.x;
  if (task >= 4) return;
  const int* v = (task < 2) ? chord : ann;
  int* t = (task < 2) ? tC : tA;
  int b = task & 1;
  v += b * S_;
  t += b * S_;
  int seg = 0, prev = 0;
  for (int s = 0; s < S_; ++s) {
    int cur = v[s];
    if (s > 0 && cur != prev) seg = s;
    prev = cur;
    t[s] = s - seg;
  }
}

// ---------------------------------------------------------------------------
// Phase 0.5: fill the sinusoidal encode matrix enc[row, d] for every table
// row. One thread per element; accurate libm powf/sin/cos, fully parallel.
// ang[d] = base^(-2*((d+1)/2)/128); even d -> sin, odd d -> cos.
// ---------------------------------------------------------------------------
__global__ void phase_enc(float* __restrict__ enc) {
  int tid = blockIdx.x * blockDim.x + threadIdx.x;
  if (tid >= ROWS_TOTAL * 128) return;
  int row = tid >> 7;
  int d = tid & 127;
  float base, v;
  if (row < ROWS_A) {
    base = 10001.0f; v = (float)(row - 23);
  } else if (row < ROWS_A + ROWS_B) {
    base = 11339.0f; v = (float)(row - ROWS_A - 15);
  } else if (row < ROWS_A + ROWS_B + ROWS_C) {
    base = 9919.0f;  v = (float)(row - (ROWS_A + ROWS_B) - 383);
  } else {
    base = 24149.0f; v = (float)(row - (ROWS_A + ROWS_B + ROWS_C) - 383);
  }
  float ang = powf(base, -2.0f * (float)((d + 1) >> 1) / 128.0f);
  float a = v * ang;
  enc[tid] = (d & 1) ? cosf(a) : sinf(a);
}

// ---------------------------------------------------------------------------
// Phase 1: tab[row, o] = b[o] + sum_d enc[row, d] * W[o, d]
// One wave = one 16x16 output tile via V_WMMA_F32_16X16X4_F32; K-loop fully
// unrolled (32 WMMAs). A-fragments stream from the enc matrix, B-fragments
// from W^T, both as 8-byte vector loads (k0+kHalf is always even).
// M-tiles: 101 total (3 A | 2 B | 48 C | 48 D) x 8 N-tiles = 808 waves.
// ---------------------------------------------------------------------------
__global__ void phase1_gemm(const float* __restrict__ W0, const float* __restrict__ b0,
                            const float* __restrict__ W1, const float* __restrict__ b1,
                            const float* __restrict__ W2, const float* __restrict__ b2,
                            const float* __restrict__ W3, const float* __restrict__ b3,
                            const float* __restrict__ enc, float* __restrict__ tab) {
  int wave = (int)((blockIdx.x * blockDim.x + threadIdx.x) >> 5);
  int lane = threadIdx.x & 31;
  if (wave >= 101 * 8) return;
  int mt = wave >> 3;  // global M-tile 0..100  (maps directly to stacked rows)
  int nt = wave & 7;   // N-tile 0..7

  const float* W;
  const float* bias;
  if (mt < 3)       { W = W0; bias = b0; }
  else if (mt < 5)  { W = W1; bias = b1; }
  else if (mt < 53) { W = W2; bias = b2; }
  else              { W = W3; bias = b3; }

  int M0 = mt * 16, N0 = nt * 16;
  int rowInLane = lane & 15;    // A: matrix row (M); B: column (N)
  int kHalf = (lane >> 4) * 2;  // lanes 0-15: K=k0+{0,1}; lanes 16-31: K=k0+{2,3}

  const float* encRow = enc + (long long)(M0 + rowInLane) * 128 + kHalf;
  const float* Wrow   = W   + (long long)(N0 + rowInLane) * 128 + kHalf;

  v8f c = {};
#pragma unroll
  for (int k0 = 0; k0 < 128; k0 += 4) {
    // A fragment: 16x4 f32 (lane holds row M=lane%16, VGPR0/1 = K pair)
    v2f a = *(const v2f*)(encRow + k0);
    // B fragment: 4x16 f32, B[k][n] = W[(N0+n)*128 + (k0+k)]
    v2f bf = *(const v2f*)(Wrow + k0);
    c = __builtin_amdgcn_wmma_f32_16x16x4_f32(
        /*neg_a=*/false, a, /*neg_b=*/false, bf,
        /*c_mod=*/(short)0, c, /*reuse_a=*/false, /*reuse_b=*/false);
  }

  // C/D layout: lane L, VGPR r -> M = r + 8*(L/16), N = L%16
  int N = N0 + rowInLane;
  float bv = bias[N];
  int Mbase = M0 + 8 * (lane >> 4);
#pragma unroll
  for (int r = 0; r < 8; ++r) {
    tab[(Mbase + r) * 128 + N] = c[r] + bv;
  }
}

// ---------------------------------------------------------------------------
// Phase 2: gather rows from L2-resident tables and stream 906 MB of output.
// One wave = one 128-float output row (v4f per lane). Non-temporal stores
// keep the streaming output from evicting the tables.
// ---------------------------------------------------------------------------
__global__ void phase2_gather(const int* __restrict__ chord, const int* __restrict__ ann,
                              const int* __restrict__ tC, const int* __restrict__ tA,
                              const float* __restrict__ ws, float* __restrict__ out) {
  const long long ROWS_PER_T = (long long)B_ * S_ * S_;
  int lane = threadIdx.x & 31;
  long long row = (long long)blockIdx.x * (blockDim.x >> 5) + (threadIdx.x >> 5);
  if (row >= 6 * ROWS_PER_T) return;

  int k = (int)(row / ROWS_PER_T);
  long long rem = row - (long long)k * ROWS_PER_T;
  int b = (int)(rem / (S_ * S_));
  int ij = (int)(rem % (S_ * S_));
  int i = ij / S_, j = ij % S_;
  int bi = b * S_ + i, bj = b * S_ + j;

  const float* tab;
  int idx;
  float keep = 1.0f;
  switch (k) {
    case 0:
      tab = ws + TABA_OFF; idx = chord[bi] - chord[bj] + 23; break;
    case 1:
      tab = ws + TABB_OFF; idx = ann[bi] - ann[bj] + 15; break;
    case 2:
    case 3:
      tab = ws + TABC_OFF; idx = tC[bi] - tC[bj] + 383;
      keep = (chord[bi] == chord[bj]) ? 1.0f : 0.0f; break;
    case 4:
      tab = ws + TABD_OFF; idx = tA[bi] + 383;   // t_minus_t0 broadcast = t[i]
      keep = (ann[bi] == ann[bj]) ? 1.0f : 0.0f; break;
    default:
      tab = ws + TABD_OFF; idx = tA[bi] - tA[bj] + 383;
      keep = (ann[bi] == ann[bj]) ? 1.0f : 0.0f; break;
  }

  v4f val = *(const v4f*)(tab + (long long)idx * 128 + lane * 4);
  val *= keep;
  v4f* dst = (v4f*)(out + row * 128 + lane * 4);
  __builtin_nontemporal_store(val, dst);
}

extern "C" void kernel_launch(void* const* d_in, const int* in_sizes, int n_in,
                              void* d_out, int out_size, void* d_ws, size_t ws_size,
                              hipStream_t stream) {
  const int* chord = (const int*)d_in[0];   // [B,S,1]
  const int* ann   = (const int*)d_in[1];   // [B,S,1]
  const float* W0 = (const float*)d_in[2];
  const float* b0 = (const float*)d_in[3];
  const float* W1 = (const float*)d_in[4];
  const float* b1 = (const float*)d_in[5];
  const float* W2 = (const float*)d_in[6];
  const float* b2 = (const float*)d_in[7];
  const float* W3 = (const float*)d_in[8];
  const float* b3 = (const float*)d_in[9];

  float* ws = (float*)d_ws;
  float* enc = ws + ENC_OFF;
  int* tC = (int*)(ws + INTS_OFF);
  int* tA = tC + B_ * S_;
  float* out = (float*)d_out;

  // Phase 0: segment-relative times (4 serial tasks)
  phase0_times<<<1, 32, 0, stream>>>(chord, ann, tC, tA);

  // Phase 0.5: encode matrix, 1616*128 elements, 1 thread each.
  phase_enc<<<(ROWS_TOTAL * 128 + 255) / 256, 256, 0, stream>>>(enc);

  // Phase 1: WMMA table GEMM. 101 M-tiles * 8 N-tiles = 808 waves.
  phase1_gemm<<<101, 256, 0, stream>>>(W0, b0, W1, b1, W2, b2, W3, b3, enc, ws);

  // Phase 2: gather + streaming write. rows = 6*B*S*S = 1,769,472; 8 rows/block.
  long long totalRows = 6LL * B_ * S_ * S_;
  int blocks = (int)((totalRows + 7) / 8);
  phase2_gather<<<blocks, 256, 0, stream>>>(chord, ann, tC, tA, ws, out);
}